// DMoN_8993661518505
// MI455X (gfx1250) — compile-verified
//
#include <hip/hip_runtime.h>
#include <hip/hip_bf16.h>

typedef __attribute__((ext_vector_type(2))) float v2f;
typedef __attribute__((ext_vector_type(8))) float v8f;

#define D 512
#define K 64

__device__ __forceinline__ v8f wmma4(v2f a, v2f b, v8f c) {
  // D = A(16x4,f32) * B(4x16,f32) + C(16x16,f32)
  return __builtin_amdgcn_wmma_f32_16x16x4_f32(false, a, false, b, (short)0, c,
                                               false, false);
}

// ---------------- Kernel 1: S = softmax(X @ W + b), row-wise -----------------
// block = 256 threads (8 waves); each wave computes a 16x64 logit tile via WMMA
__global__ void k1_gemm_softmax(const float* __restrict__ X,
                                const float* __restrict__ W,
                                const float* __restrict__ bias,
                                float* __restrict__ S) {
  extern __shared__ float smem[];
  float* lw = smem;  // [D*K] W cache
  const int tid = threadIdx.x;
  for (int i = tid; i < D * K; i += blockDim.x) lw[i] = W[i];
  __syncthreads();

  const int wave = tid >> 5;
  const int lane = tid & 31;
  const int half = lane >> 4;
  const int l16 = lane & 15;
  float* tile = smem + D * K + wave * (16 * 65);  // padded 16x64 tile

  const int row_base = (blockIdx.x * 8 + wave) * 16;
  const int k0 = half ? 2 : 0;
  const int k1o = half ? 3 : 1;

  v8f acc[4] = {};
  const float* xrow = X + (size_t)(row_base + l16) * D;
  for (int kb = 0; kb < D; kb += 4) {
    v2f a;
    a.x = xrow[kb + k0];
    a.y = xrow[kb + k1o];
#pragma unroll
    for (int nt = 0; nt < 4; ++nt) {
      v2f b;
      b.x = lw[(kb + k0) * K + nt * 16 + l16];
      b.y = lw[(kb + k1o) * K + nt * 16 + l16];
      acc[nt] = wmma4(a, b, acc[nt]);
    }
  }
  // write logits (+bias) to LDS tile: element (M,N): M=j+8*half, N=nt*16+l16
#pragma unroll
  for (int nt = 0; nt < 4; ++nt) {
    float bv = bias[nt * 16 + l16];
#pragma unroll
    for (int j = 0; j < 8; ++j) {
      int m = j + (half ? 8 : 0);
      tile[m * 65 + nt * 16 + l16] = acc[nt][j] + bv;
    }
  }
  __syncthreads();
  if (lane < 16) {  // one row per lane
    float* r = tile + lane * 65;
    float mx = r[0];
    for (int c = 1; c < K; ++c) mx = fmaxf(mx, r[c]);
    float sum = 0.f;
    for (int c = 0; c < K; ++c) {
      float e = __expf(r[c] - mx);
      r[c] = e;
      sum += e;
    }
    float inv = 1.f / sum;
    for (int c = 0; c < K; ++c) r[c] *= inv;
  }
  __syncthreads();
  float* sout = S + (size_t)row_base * K;
  for (int idx = lane; idx < 16 * K; idx += 32) {
    int m = idx >> 6, c = idx & 63;
    sout[idx] = tile[m * 65 + c];
  }
}

// ---------------- Kernel 2: degrees + m ----------------
__global__ void k2_degrees(const int* __restrict__ esrc,
                           const int* __restrict__ edst,
                           const float* __restrict__ ev, int E,
                           float* __restrict__ deg_row,
                           float* __restrict__ deg_col,
                           float* __restrict__ scal) {
  int e = blockIdx.x * blockDim.x + threadIdx.x;
  float v = 0.f;
  if (e < E) {
    v = ev[e];
    atomicAdd(deg_row + esrc[e], v);
    atomicAdd(deg_col + edst[e], v);
  }
  for (int off = 16; off > 0; off >>= 1) v += __shfl_down(v, off, 32);
  if ((threadIdx.x & 31) == 0) atomicAdd(scal + 0, v);  // m
}

// ---------------- Kernel 3: AS[s,:] += v * S[t,:] ----------------
__global__ void k3_AS(const int* __restrict__ esrc, const int* __restrict__ edst,
                      const float* __restrict__ ev, const float* __restrict__ S,
                      float* __restrict__ AS, int E) {
  int gid = blockIdx.x * blockDim.x + threadIdx.x;
  int e = gid >> 6;
  int c = gid & 63;
  if (e >= E) return;
  int s = esrc[e], t = edst[e];
  float v = ev[e];
  atomicAdd(AS + (size_t)s * K + c, v * S[(size_t)t * K + c]);
}

// ---------------- Kernel 4: row reductions (cs, dtS, tr(S^T A S), StLS) ------
__global__ void k4_rowstats(const float* __restrict__ S,
                            const float* __restrict__ AS,
                            const float* __restrict__ deg_row,
                            const float* __restrict__ deg_col,
                            float* __restrict__ cs, float* __restrict__ dtS,
                            float* __restrict__ StLS, float* __restrict__ scal,
                            int rows_per_wg) {
  __shared__ float sS[K];
  __shared__ float sLS[K];
  const int tid = threadIdx.x;       // 256 threads
  const int c_own = tid >> 2;        // StLS row owned by this thread
  const int c2b = (tid & 3) << 4;    // StLS col range [c2b, c2b+16)
  float acc[16] = {};
  float cs_p = 0.f, dts_p = 0.f, tr_p = 0.f;
  const int base = blockIdx.x * rows_per_wg;
  for (int r = 0; r < rows_per_wg; ++r) {
    int i = base + r;
    if (tid < K) {
      float sv = S[(size_t)i * K + tid];
      float av = AS[(size_t)i * K + tid];
      sS[tid] = sv;
      sLS[tid] = deg_row[i] * sv - av;  // (L S)[i, tid]
      cs_p += sv;
      dts_p += deg_col[i] * sv;
      tr_p += av * sv;
    }
    __syncthreads();
    float lsc = sLS[c_own];
#pragma unroll
    for (int e = 0; e < 16; ++e) acc[e] += lsc * sS[c2b + e];
    __syncthreads();
  }
#pragma unroll
  for (int e = 0; e < 16; ++e) atomicAdd(StLS + c_own * K + c2b + e, acc[e]);
  if (tid < K) {
    atomicAdd(cs + tid, cs_p);
    atomicAdd(dtS + tid, dts_p);
  }
  for (int off = 16; off > 0; off >>= 1) tr_p += __shfl_down(tr_p, off, 32);
  if ((tid & 31) == 0) atomicAdd(scal + 1, tr_p);  // trace(S^T A S)
}

// ---------------- Kernel 5: T = S^T @ X (split-K WMMA + atomic merge) --------
// block = 1024 threads (32 waves): wave -> (c-tile, d-group); K-chunk per block
__global__ void k5_STX(const float* __restrict__ S, const float* __restrict__ X,
                       float* __restrict__ T, int rows_per_wg) {
  const int tid = threadIdx.x;
  const int wave = tid >> 5, lane = tid & 31;
  const int half = lane >> 4, l16 = lane & 15;
  const int ct = wave & 3;   // c tile (16 rows of T)
  const int dg = wave >> 2;  // d group of 64
  const int kbase = blockIdx.x * rows_per_wg;
  const int k0 = half ? 2 : 0, k1o = half ? 3 : 1;
  v8f acc[4] = {};
  for (int kb = 0; kb < rows_per_wg; kb += 4) {
    size_t i0 = (size_t)(kbase + kb + k0);
    size_t i1 = (size_t)(kbase + kb + k1o);
    v2f a;  // A[M=c, K=i] = S[i, c]
    a.x = S[i0 * K + ct * 16 + l16];
    a.y = S[i1 * K + ct * 16 + l16];
#pragma unroll
    for (int nt = 0; nt < 4; ++nt) {
      v2f b;  // B[K=i, N=d] = X[i, d]
      b.x = X[i0 * D + dg * 64 + nt * 16 + l16];
      b.y = X[i1 * D + dg * 64 + nt * 16 + l16];
      acc[nt] = wmma4(a, b, acc[nt]);
    }
  }
#pragma unroll
  for (int nt = 0; nt < 4; ++nt)
#pragma unroll
    for (int j = 0; j < 8; ++j) {
      int c = ct * 16 + j + (half ? 8 : 0);
      int d = dg * 64 + nt * 16 + l16;
      atomicAdd(T + c * D + d, acc[nt][j]);
    }
}

// ---------------- Kernel 5b: FP = selu(T / cluster_sizes) ----------------
__global__ void k5b_selu(const float* __restrict__ T, const float* __restrict__ cs,
                         float* __restrict__ FP) {
  int idx = blockIdx.x * blockDim.x + threadIdx.x;
  if (idx >= K * D) return;
  int c = idx / D;
  float x = T[idx] / cs[c];
  const float alpha = 1.6732632423543772f;
  const float scale = 1.0507009873554805f;
  FP[idx] = scale * (x > 0.f ? x : alpha * (__expf(x) - 1.f));
}

// ---------------- Kernel 6: frob = ||S @ FP - X||_F^2 (WMMA) ----------------
// block = 256 (8 waves); FP cached in LDS; each wave a 16-row tile, loops d
__global__ void k6_frob(const float* __restrict__ S, const float* __restrict__ FP,
                        const float* __restrict__ X, float* __restrict__ scal) {
  extern __shared__ float smem[];
  float* lfp = smem;  // [K*D]
  const int tid = threadIdx.x;
  for (int i = tid; i < K * D; i += blockDim.x) lfp[i] = FP[i];
  const int wave = tid >> 5, lane = tid & 31;
  const int half = lane >> 4, l16 = lane & 15;
  float* stile = smem + K * D + wave * (16 * 65);
  const int rb = (blockIdx.x * 8 + wave) * 16;
  for (int idx = lane; idx < 16 * K; idx += 32) {
    int m = idx >> 6, c = idx & 63;
    stile[m * 65 + c] = S[(size_t)(rb + m) * K + c];
  }
  __syncthreads();
  const int k0 = half ? 2 : 0, k1o = half ? 3 : 1;
  float sq = 0.f;
  for (int dg = 0; dg < 8; ++dg) {
    v8f acc[4] = {};
    for (int kb = 0; kb < K; kb += 4) {
      v2f a;  // A[M=i, K=c] = S[i, c]
      a.x = stile[l16 * 65 + kb + k0];
      a.y = stile[l16 * 65 + kb + k1o];
#pragma unroll
      for (int nt = 0; nt < 4; ++nt) {
        v2f b;  // B[K=c, N=d] = FP[c, d]
        b.x = lfp[(kb + k0) * D + dg * 64 + nt * 16 + l16];
        b.y = lfp[(kb + k1o) * D + dg * 64 + nt * 16 + l16];
        acc[nt] = wmma4(a, b, acc[nt]);
      }
    }
#pragma unroll
    for (int nt = 0; nt < 4; ++nt)
#pragma unroll
      for (int j = 0; j < 8; ++j) {
        int i = rb + j + (half ? 8 : 0);
        int d = dg * 64 + nt * 16 + l16;
        float r = acc[nt][j] - X[(size_t)i * D + d];
        sq += r * r;
      }
  }
  for (int off = 16; off > 0; off >>= 1) sq += __shfl_down(sq, off, 32);
  if (lane == 0) atomicAdd(scal + 2, sq);
}

// ---------------- Kernel 7: M2 = FP@FP^T; trace_loss = sum LS[:64]*M2^T -----
__global__ void k7_trace(const float* __restrict__ FP, const float* __restrict__ S,
                         const float* __restrict__ AS,
                         const float* __restrict__ deg_row,
                         float* __restrict__ scal) {
  __shared__ float m2[K * K];
  const int tid = threadIdx.x;  // 256
  const int c = tid >> 2, ib = (tid & 3) << 4;
  for (int e = 0; e < 16; ++e) {
    int i = ib + e;
    const float* a = FP + c * D;
    const float* b = FP + i * D;
    float s = 0.f;
    for (int d = 0; d < D; ++d) s += a[d] * b[d];
    m2[c * K + i] = s;
  }
  __syncthreads();
  float t = 0.f;
  for (int e = 0; e < 16; ++e) {
    int idx = tid * 16 + e;
    int i = idx >> 6, cc = idx & 63;
    float ls = deg_row[i] * S[i * K + cc] - AS[i * K + cc];
    t += ls * m2[cc * K + i];
  }
  for (int off = 16; off > 0; off >>= 1) t += __shfl_down(t, off, 32);
  if ((tid & 31) == 0) atomicAdd(scal + 3, t);
}

// ---------------- Kernel 8: logabsdet(StLS + ones) via LU, 64 threads -------
__global__ void k8_slogdet(const float* __restrict__ StLS, float* __restrict__ scal) {
  __shared__ float G[K * K];
  __shared__ int piv;
  const int tid = threadIdx.x;  // 64
  for (int i = tid; i < K * K; i += 64) G[i] = StLS[i] + 1.0f;
  __syncthreads();
  float lad = 0.f;
  for (int k = 0; k < K; ++k) {
    if (tid == 0) {
      int p = k;
      float best = fabsf(G[k * K + k]);
      for (int i = k + 1; i < K; ++i) {
        float v = fabsf(G[i * K + k]);
        if (v > best) { best = v; p = i; }
      }
      piv = p;
    }
    __syncthreads();
    int p = piv;
    if (p != k) {
      float a = G[k * K + tid], b = G[p * K + tid];
      G[k * K + tid] = b;
      G[p * K + tid] = a;
    }
    __syncthreads();
    float pv = G[k * K + k];
    if (tid == 0) lad += logf(fabsf(pv));
    if (tid > k) {
      float f = G[tid * K + k] / pv;
      for (int c = k; c < K; ++c) G[tid * K + c] -= f * G[k * K + c];
    }
    __syncthreads();
  }
  if (tid == 0) scal[4] = lad;
}

// ---------------- Kernel 9: assemble total loss ----------------
__global__ void k9_final(const float* __restrict__ scal, const float* __restrict__ cs,
                         const float* __restrict__ dtS, float* __restrict__ loss,
                         int n) {
  float m = scal[0], trAS = scal[1], frob_sq = scal[2];
  float trace_loss = scal[3], lad = scal[4];
  float sq_dts = 0.f, cs2 = 0.f;
  for (int c = 0; c < K; ++c) {
    sq_dts += dtS[c] * dtS[c];
    cs2 += cs[c] * cs[c];
  }
  float cs_norm = sqrtf(cs2);
  float spectral = -(trAS - sq_dts / (2.f * m)) / (2.f * m);
  float collapse = cs_norm / (float)n * sqrtf((float)K) - 1.f;
  float smooth = lad * -0.0001f;                 // H_SMOOTH
  float frob = sqrtf(frob_sq) * 0.0001f;         // H_ALPHA
  float one_two = cs_norm * 0.0001f;             // H_LAMBDA (S >= 0)
  loss[0] = spectral + 0.1f * collapse + smooth + trace_loss + frob + one_two;
}

extern "C" void kernel_launch(void* const* d_in, const int* in_sizes, int n_in,
                              void* d_out, int out_size, void* d_ws, size_t ws_size,
                              hipStream_t stream) {
  (void)n_in; (void)ws_size;
  const float* X = (const float*)d_in[0];
  const int* esrc = (const int*)d_in[1];
  const int* edst = (const int*)d_in[2];
  const float* ev = (const float*)d_in[3];
  const float* W = (const float*)d_in[4];
  const float* bias = (const float*)d_in[5];

  const int n = in_sizes[0] / D;  // 65536
  const int E = in_sizes[1];      // 2097152 (already symmetrized)

  float* ws = (float*)d_ws;
  size_t off = 0;
  float* AS = ws + off;      off += (size_t)n * K;
  float* deg_row = ws + off; off += n;
  float* deg_col = ws + off; off += n;
  float* cs = ws + off;      off += K;
  float* dtS = ws + off;     off += K;
  float* T = ws + off;       off += (size_t)K * D;
  float* StLS = ws + off;    off += (size_t)K * K;
  float* scal = ws + off;    off += 16;
  hipMemsetAsync(d_ws, 0, off * sizeof(float), stream);

  float* FP = (float*)d_out;
  float* S = (float*)d_out + (size_t)K * D;
  float* loss = (float*)d_out + (size_t)out_size - 1;

  size_t sh1 = (size_t)(D * K + 8 * 16 * 65) * sizeof(float);
  k1_gemm_softmax<<<n / 128, 256, sh1, stream>>>(X, W, bias, S);
  k2_degrees<<<(E + 255) / 256, 256, 0, stream>>>(esrc, edst, ev, E, deg_row,
                                                  deg_col, scal);
  k3_AS<<<((size_t)E * 64 + 255) / 256, 256, 0, stream>>>(esrc, edst, ev, S, AS, E);
  k4_rowstats<<<n / 256, 256, 0, stream>>>(S, AS, deg_row, deg_col, cs, dtS,
                                           StLS, scal, 256);
  k5_STX<<<n / 512, 1024, 0, stream>>>(S, X, T, 512);
  k5b_selu<<<(K * D) / 256, 256, 0, stream>>>(T, cs, FP);
  size_t sh6 = (size_t)(K * D + 8 * 16 * 65) * sizeof(float);
  k6_frob<<<n / 128, 256, sh6, stream>>>(S, FP, X, scal);
  k7_trace<<<1, 256, 0, stream>>>(FP, S, AS, deg_row, scal);
  k8_slogdet<<<1, 64, 0, stream>>>(StLS, scal);
  k9_final<<<1, 1, 0, stream>>>(scal, cs, dtS, loss, n);
}